// Encoder_57380763074770
// MI455X (gfx1250) — compile-verified
//
#include <hip/hip_runtime.h>
#include <math.h>

// ---------------------------------------------------------------------------
// Problem constants (from the reference)
// ---------------------------------------------------------------------------
#define H      128      // hidden size
#define F_IN   33       // GRU input features
#define KPAD   36       // K padded to multiple of 4 for 16x16x4 WMMA
#define STEPS  64       // independent rows
#define G3H    384      // 3*H

typedef __attribute__((ext_vector_type(2))) float v2f;
typedef __attribute__((ext_vector_type(8))) float v8f;

__device__ __forceinline__ float sigmoidf_(float x) {
    return 1.0f / (1.0f + __expf(-x));
}

// ---------------------------------------------------------------------------
// Kernel 1: build xi [64 x 36] (zero-padded) and gh [384] = h0 @ W_hh^T + b_hh
// One block, 384 threads. Tiny, latency-bound prep work.
// Workspace layout (floats): ws[0 .. 2303]  = xi (row-major, 64x36)
//                            ws[2304..2687] = gh (384)
// ---------------------------------------------------------------------------
__global__ void gru_prep_kernel(const float* __restrict__ x,
                                const int*   __restrict__ ip,
                                const int*   __restrict__ port,
                                const float* __restrict__ h0,       // hidden[0][0]
                                const float* __restrict__ ip_emb,   // [256,1]
                                const float* __restrict__ port_emb, // [70000,4]
                                const float* __restrict__ W_hh,     // [384,128]
                                const float* __restrict__ b_hh,     // [384]
                                float*       __restrict__ ws)
{
    const int tid = threadIdx.x;           // 0..383
    float* xi = ws;
    float* gh = ws + STEPS * KPAD;

    // --- gh[tid] = b_hh[tid] + dot(h0, W_hh[tid,:]) ---
    {
        float acc = b_hh[tid];
        const float* wrow = W_hh + tid * H;
        #pragma unroll 8
        for (int i = 0; i < H; ++i) acc = fmaf(h0[i], wrow[i], acc);
        gh[tid] = acc;
    }

    // --- xi gather/concat: [x(17) | ip_emb gather(8) | port_emb gather(8) | pad(3)] ---
    for (int idx = tid; idx < STEPS * KPAD; idx += G3H) {
        const int row = idx / KPAD;
        const int col = idx - row * KPAD;
        float v;
        if (col < 17) {
            v = x[row * 17 + col];
        } else if (col < 25) {
            const int e = ip[row * 8 + (col - 17)];
            v = ip_emb[e];                       // emb dim 1
        } else if (col < F_IN) {
            const int j = col - 25;              // 0..7
            const int p = port[row * 2 + (j >> 2)];
            v = port_emb[p * 4 + (j & 3)];       // emb dim 4
        } else {
            v = 0.0f;                            // K padding
        }
        xi[idx] = v;
    }
}

// ---------------------------------------------------------------------------
// Kernel 2: gi = xi @ W_ih^T via V_WMMA_F32_16X16X4_F32, fused GRU gate math.
// Grid: 4 blocks (M tiles of 16 rows), 256 threads = 8 waves (N tiles of 16
// hidden columns). Each wave owns 3 accumulators (r/z/n gates) and issues
// 9 K-steps x 3 = 27 wmma ops. EXEC is all-ones throughout (no divergence).
//
// Fragment layouts (ISA 7.12.2, wave32):
//   A (16x4 f32, 2 VGPR): lane&15 = M; lane>=16 -> K+=2; v0=K, v1=K+1
//   B (4x16 f32, 2 VGPR): lane&15 = N; lane>=16 -> K+=2; v0=K, v1=K+1
//   C (16x16 f32, 8 VGPR): vgpr v -> M = v + (lane>=16 ? 8:0); N = lane&15
// ---------------------------------------------------------------------------
__global__ void gru_wmma_gate_kernel(const float* __restrict__ ws,   // xi + gh
                                     const float* __restrict__ W_ih, // [384,33]
                                     const float* __restrict__ b_ih, // [384]
                                     const float* __restrict__ h0,   // [128]
                                     float*       __restrict__ out)  // [64*128 + 128]
{
    const float* xi = ws;
    const float* gh = ws + STEPS * KPAD;

    const int lane  = threadIdx.x & 31;
    const int wave  = threadIdx.x >> 5;        // 0..7  -> n tile
    const int m0    = blockIdx.x << 4;         // 0,16,32,48
    const int n0    = wave << 4;               // 0..112
    const int laneN = lane & 15;               // N (for B/C) or M (for A)
    const int koff  = (lane >> 4) << 1;        // 0 or 2: K offset by lane half
    const int mhalf = (lane >> 4) << 3;        // 0 or 8: M offset for C vgprs

    const int col  = n0 + laneN;               // output hidden column 0..127
    const int colr = col;                      // r-gate row of W_ih
    const int colz = col + H;                  // z-gate
    const int coln = col + 2 * H;              // n-gate

    v8f cr = {}; v8f cz = {}; v8f cn = {};

    const float* arow = xi + (m0 + laneN) * KPAD;   // A: M = laneN
    const float* wr = W_ih + colr * F_IN;
    const float* wz = W_ih + colz * F_IN;
    const float* wn = W_ih + coln * F_IN;

    #pragma unroll
    for (int k0 = 0; k0 < KPAD; k0 += 4) {
        const int kA = k0 + koff;
        const int kB = kA + 1;

        v2f a;
        a.x = arow[kA];                        // xi padded -> no guard needed
        a.y = arow[kB];

        // W_ih has only 33 K entries; mask the pad region with selects
        const bool okA = (kA < F_IN);
        const bool okB = (kB < F_IN);

        v2f br, bz, bn;
        br.x = okA ? wr[kA] : 0.0f;  br.y = okB ? wr[kB] : 0.0f;
        bz.x = okA ? wz[kA] : 0.0f;  bz.y = okB ? wz[kB] : 0.0f;
        bn.x = okA ? wn[kA] : 0.0f;  bn.y = okB ? wn[kB] : 0.0f;

        cr = __builtin_amdgcn_wmma_f32_16x16x4_f32(false, a, false, br,
                                                   (short)0, cr, false, false);
        cz = __builtin_amdgcn_wmma_f32_16x16x4_f32(false, a, false, bz,
                                                   (short)0, cz, false, false);
        cn = __builtin_amdgcn_wmma_f32_16x16x4_f32(false, a, false, bn,
                                                   (short)0, cn, false, false);
    }

    // Per-lane (per hidden column) scalars
    const float bihr = b_ih[colr], bihz = b_ih[colz], bihn = b_ih[coln];
    const float ghr  = gh[colr],   ghz  = gh[colz],   ghn  = gh[coln];
    const float h0c  = h0[col];

    #pragma unroll
    for (int v = 0; v < 8; ++v) {
        const int m = m0 + v + mhalf;          // output row 0..63
        const float r = sigmoidf_(cr[v] + bihr + ghr);
        const float z = sigmoidf_(cz[v] + bihz + ghz);
        const float n = tanhf(cn[v] + bihn + r * ghn);
        const float o = (1.0f - z) * n + z * h0c;
        out[m * H + col] = o;
        if (m == STEPS - 1) out[STEPS * H + col] = o;  // new_hidden
    }
}

// ---------------------------------------------------------------------------
// Launch
// ---------------------------------------------------------------------------
extern "C" void kernel_launch(void* const* d_in, const int* in_sizes, int n_in,
                              void* d_out, int out_size, void* d_ws, size_t ws_size,
                              hipStream_t stream)
{
    const float* x        = (const float*)d_in[0];
    const int*   ip       = (const int*)  d_in[1];
    const int*   port     = (const int*)  d_in[2];
    const float* hidden   = (const float*)d_in[3];   // [1,1,128]
    const float* ip_emb   = (const float*)d_in[4];
    const float* port_emb = (const float*)d_in[5];
    const float* W_ih     = (const float*)d_in[6];
    const float* W_hh     = (const float*)d_in[7];
    const float* b_ih     = (const float*)d_in[8];
    const float* b_hh     = (const float*)d_in[9];
    float*       out      = (float*)d_out;
    float*       ws       = (float*)d_ws;            // 64*36 + 384 floats

    gru_prep_kernel<<<1, G3H, 0, stream>>>(x, ip, port, hidden,
                                           ip_emb, port_emb, W_hh, b_hh, ws);

    gru_wmma_gate_kernel<<<4, 256, 0, stream>>>(ws, W_ih, b_ih, hidden, out);
}